// RPN_48129403519607
// MI455X (gfx1250) — compile-verified
//
#include <hip/hip_runtime.h>
#include <hip/hip_bf16.h>

#define N_BOXES 4000
#define NPAD    4096
#define K_TOP   1000
#define NMS_TH  0.7f

typedef unsigned int u32x4 __attribute__((ext_vector_type(4)));
typedef int          i32x4 __attribute__((ext_vector_type(4)));
typedef int          i32x8 __attribute__((ext_vector_type(8)));

// ---------------------------------------------------------------------------
// Kernel 1: per-image bitonic sort by score (desc) + gather sorted boxes into
//           workspace tile [B][NPAD][4].
// ---------------------------------------------------------------------------
__global__ void __launch_bounds__(256)
rpn_sort_kernel(const float* __restrict__ boxes,   // [B][N][4]
                const float* __restrict__ scores,  // [B][N]
                float* __restrict__ sorted_boxes)  // [B][NPAD][4]
{
    const int b   = blockIdx.x;
    const int tid = threadIdx.x;
    const int nt  = blockDim.x;

    __shared__ float s_key[NPAD];
    __shared__ int   s_idx[NPAD];

    const float* sc = scores + (size_t)b * N_BOXES;
    const float* bx = boxes  + (size_t)b * N_BOXES * 4;

    // Warm L2 for the gather at the end (gfx1250: global_prefetch_b8).
    for (int t = tid * 64; t < N_BOXES * 4; t += nt * 64)
        __builtin_prefetch(bx + t, 0, 1);

    // key = -score => ascending bitonic sort yields descending score order.
    for (int t = tid; t < NPAD; t += nt) {
        s_key[t] = (t < N_BOXES) ? -sc[t] : 3.4e38f;
        s_idx[t] = t;
    }
    __syncthreads();

    for (int k = 2; k <= NPAD; k <<= 1) {
        for (int j = k >> 1; j > 0; j >>= 1) {
            for (int t = tid; t < NPAD; t += nt) {
                int ixj = t ^ j;
                if (ixj > t) {
                    float a = s_key[t], c = s_key[ixj];
                    bool up = ((t & k) == 0);
                    if (up ? (a > c) : (a < c)) {
                        s_key[t] = c; s_key[ixj] = a;
                        int ia = s_idx[t]; s_idx[t] = s_idx[ixj]; s_idx[ixj] = ia;
                    }
                }
            }
            __syncthreads();
        }
    }

    // Gather boxes in sorted order (128-bit loads/stores).
    float* ob = sorted_boxes + (size_t)b * NPAD * 4;
    for (int t = tid; t < NPAD; t += nt) {
        int g = s_idx[t];
        float4 v;
        if (g < N_BOXES) v = ((const float4*)bx)[g];
        else             v = make_float4(0.f, 0.f, 0.f, 0.f);
        ((float4*)ob)[t] = v;
    }
}

// ---------------------------------------------------------------------------
// Kernel 2: per-image greedy NMS on the LDS-resident sorted tile.
//           Tile staged with the Tensor Data Mover (async DMA to LDS).
//           One barrier round per KEPT box (<=1000), not per list position.
// ---------------------------------------------------------------------------
__global__ void __launch_bounds__(256)
rpn_nms_kernel(const float* __restrict__ sorted_boxes, // [B][NPAD][4]
               float* __restrict__ out)                // [B][K_TOP][5]
{
    const int b   = blockIdx.x;
    const int tid = threadIdx.x;
    const int nt  = blockDim.x;

    __shared__ __align__(16) float s_box[NPAD * 4];   // 64 KB (TDM destination)
    __shared__ float         s_area[NPAD];            // 16 KB
    __shared__ unsigned char s_alive[NPAD];           //  4 KB
    __shared__ int           s_pos;                   // scan cursor (thread 0)
    __shared__ int           s_cur;                   // broadcast: kept index or -1

    const float* src = sorted_boxes + (size_t)b * NPAD * 4;

    if (tid == 0) {
        s_pos = 0;

        // ---- Tensor DMA descriptor: 16384 f32 contiguous -> LDS s_box ----
        unsigned long long ga = (unsigned long long)(size_t)src;
        u32x4 g0;
        g0.x = 1u;                                      // count=1 (valid D#)
        g0.y = (unsigned)(size_t)&s_box[0];             // lds_addr (low 32 of flat)
        g0.z = (unsigned)(ga & 0xffffffffu);            // global_addr[31:0]
        g0.w = (unsigned)((ga >> 32) & 0x01ffffffu)     // global_addr[56:32]
             | 0x80000000u;                             // type=2 ("image")
        i32x8 g1;
        g1[0] = (int)(2u << 16);                        // data_size=4B, no mcast
        g1[1] = (int)(16384u << 16);                    // tensor_dim0[15:0]
        g1[2] = (int)(1u << 16);                        // dim0 hi=0, tensor_dim1=1
        g1[3] = (int)(16384u << 16);                    // tile_dim0=16384
        g1[4] = 1;                                      // tile_dim1=1, tile_dim2=0
        g1[5] = 16384;                                  // tensor_dim0_stride lo
        g1[6] = (int)(16384u << 16);                    // dim1_stride lo16
        g1[7] = 0;
        i32x4 z4 = {0, 0, 0, 0};
#if __has_include(<hip/amd_detail/amd_gfx1250_TDM.h>)
        i32x8 z8 = {0, 0, 0, 0, 0, 0, 0, 0};
        __builtin_amdgcn_tensor_load_to_lds(g0, g1, z4, z4, z8, 0);
#else
        __builtin_amdgcn_tensor_load_to_lds(g0, g1, z4, z4, 0);
#endif
        __builtin_amdgcn_s_wait_tensorcnt(0);           // s_wait_tensorcnt 0
    }
    __syncthreads();                                    // tile visible to all waves

    // Areas + alive flags; pre-fill all K output rows with (b,0,0,0,0).
    for (int t = tid; t < NPAD; t += nt) {
        float4 v = ((const float4*)s_box)[t];
        s_area[t]  = (v.z - v.x) * (v.w - v.y);
        s_alive[t] = (t < N_BOXES) ? 1 : 0;
    }
    float* out_img = out + (size_t)b * K_TOP * 5;
    for (int t = tid; t < K_TOP; t += nt) {
        float* r = out_img + (size_t)t * 5;
        r[0] = (float)b; r[1] = 0.f; r[2] = 0.f; r[3] = 0.f; r[4] = 0.f;
    }
    __syncthreads();

    // Greedy scan: each round thread 0 advances to the next live entry
    // (previous round's suppression is barrier-synchronized), emits it, and
    // all 8 waves suppress later boxes.  <= K_TOP rounds total.
    for (int r = 0; r < K_TOP; ++r) {
        if (tid == 0) {
            int i = s_pos;
            while (i < N_BOXES && !s_alive[i]) ++i;     // skip dead entries
            if (i < N_BOXES) {
                float4 bi = ((const float4*)s_box)[i];
                float* row = out_img + (size_t)r * 5;
                row[0] = (float)b;
                row[1] = bi.x; row[2] = bi.y; row[3] = bi.z; row[4] = bi.w;
                s_pos = i + 1;
                // Last slot filled: no further output depends on suppression.
                s_cur = (r + 1 < K_TOP) ? i : -1;
            } else {
                s_cur = -1;                             // list exhausted
            }
        }
        __syncthreads();
        int i = s_cur;
        if (i < 0) break;                               // uniform exit
        {
            float4 bi = ((const float4*)s_box)[i];
            float  ai = s_area[i];
            for (int j = i + 1 + tid; j < N_BOXES; j += nt) {
                if (!s_alive[j]) continue;
                float4 bj = ((const float4*)s_box)[j];
                float xx1 = fmaxf(bi.x, bj.x);
                float yy1 = fmaxf(bi.y, bj.y);
                float xx2 = fminf(bi.z, bj.z);
                float yy2 = fminf(bi.w, bj.w);
                float w = fmaxf(xx2 - xx1, 0.0f);
                float h = fmaxf(yy2 - yy1, 0.0f);
                float inter = w * h;
                float uni = ai + s_area[j] - inter + 1e-9f;
                if (inter > NMS_TH * uni)               // iou > thresh, div-free
                    s_alive[j] = 0;
            }
        }
        __syncthreads();
    }
}

// ---------------------------------------------------------------------------
extern "C" void kernel_launch(void* const* d_in, const int* in_sizes, int n_in,
                              void* d_out, int out_size, void* d_ws, size_t ws_size,
                              hipStream_t stream) {
    const float* boxes  = (const float*)d_in[0];   // [B][4000][4] f32
    const float* scores = (const float*)d_in[1];   // [B][4000]    f32
    float* out = (float*)d_out;                    // [B][1000][5] f32

    const int B = in_sizes[1] / N_BOXES;
    float* sorted_boxes = (float*)d_ws;            // B * NPAD * 4 f32 = 512 KB

    rpn_sort_kernel<<<B, 256, 0, stream>>>(boxes, scores, sorted_boxes);
    rpn_nms_kernel<<<B, 256, 0, stream>>>(sorted_boxes, out);
}